// CGP_LAYER_76751065579584
// MI455X (gfx1250) — compile-verified
//
#include <hip/hip_runtime.h>
#include <hip/hip_bf16.h>
#include <math.h>

#define NSEQ 512
#define NB 4
#define NH 8
#define VDD 64
#define NBH (NB*NH)
#define HDIM 512
#define SSAMP 4
#define JITV 1e-2f

typedef __attribute__((ext_vector_type(2))) float v2f;
typedef __attribute__((ext_vector_type(8))) float v8f;
typedef __attribute__((ext_vector_type(4))) unsigned int u32x4;
typedef __attribute__((ext_vector_type(4))) int i32x4;
typedef __attribute__((ext_vector_type(8))) int i32x8;

static __device__ __forceinline__ v8f wmma4(v2f a, v2f b, v8f c) {
  // V_WMMA_F32_16X16X4_F32 : D = A(16x4) x B(4x16) + C(16x16), fp32
  return __builtin_amdgcn_wmma_f32_16x16x4_f32(false, a, false, b, (short)0, c, false, false);
}

// ---------------------------------------------------------------------------
// TDM: DMA a 2D tile (tile_k contiguous elems x tile_m rows, f32, row stride
// lda elems) from global memory into LDS at lds_addr, adding `pad` dwords of
// LDS padding after every row (pad codes per D# spec).
// Descriptor layout per CDNA5 ISA (group0 128b, group1 256b).
// ---------------------------------------------------------------------------
static __device__ __forceinline__ void tdm_load_tile(
    unsigned lds_addr, const float* gptr, int lda_elems,
    unsigned tile_k, unsigned tile_m,
    unsigned pad_interval_code, unsigned pad_amount_code)
{
  unsigned long long ga = (unsigned long long)gptr;
  u32x4 g0;
  g0.x = 1u;                                          // count=1 valid user D#
  g0.y = lds_addr;                                    // LDS byte address
  g0.z = (unsigned)(ga & 0xFFFFFFFFu);                // global_addr[31:0]
  g0.w = (unsigned)((ga >> 32) & 0x01FFFFFFu) | (2u << 30);  // addr[56:32] | type=2
  unsigned long long st0 = (unsigned long long)(unsigned)lda_elems;
  unsigned w0 = (2u << 16)                            // data_size = 4 bytes
              | (1u << 20)                            // pad_enable
              | (pad_interval_code << 22)
              | (pad_amount_code << 25);
  i32x8 g1;
  g1[0] = (int)w0;
  g1[1] = (int)((tile_k & 0xFFFFu) << 16);            // tensor_dim0[15:0] (== tile_k)
  g1[2] = (int)((tile_k >> 16) | ((tile_m & 0xFFFFu) << 16)); // td0 hi | td1 lo
  g1[3] = (int)((tile_m >> 16) | (tile_k << 16));     // td1 hi | tile_dim0
  g1[4] = (int)tile_m;                                // tile_dim1 | tile_dim2=0
  g1[5] = (int)(st0 & 0xFFFFFFFFu);                   // tensor_dim0_stride lo32
  g1[6] = (int)((st0 >> 32) & 0xFFFFu);               // stride hi16 | dim1_stride lo16=0
  g1[7] = 0;
  i32x4 gz4 = {0, 0, 0, 0};
#if defined(__clang_major__) && (__clang_major__ >= 23)
  i32x8 gz8 = {0, 0, 0, 0, 0, 0, 0, 0};
  __builtin_amdgcn_tensor_load_to_lds(g0, g1, gz4, gz4, gz8, 0);
#else
  __builtin_amdgcn_tensor_load_to_lds(g0, g1, gz4, gz4, 0);
#endif
}

// ---------------------------------------------------------------------------
// Generic batched GEMM: C = alpha * op(A) @ op(B) + beta*C (+ bias[col]) (+ diag)
// grid: (N/64, M/64, nbh); block: 128 threads (4 waves), 64x64 tile / WG,
// each wave owns a 32x32 sub-tile = 2x2 WMMA 16x16 accumulators.
// LDS tiles are stored K-contiguous (As[m][k], Bs[n][k], row stride 34) so a
// WMMA fragment is one aligned ds_load_b64.  For !TA the A tile is staged by
// the Tensor Data Mover (wave 0 issues, s_wait_tensorcnt, then WG barrier);
// all !TA callsites have M%64==0 and K%32==0 so tiles are always full.
// ---------------------------------------------------------------------------
template<bool TA, bool TB>
__global__ __launch_bounds__(128)
void gemm_k(const float* __restrict__ A, const float* __restrict__ B,
            float* __restrict__ C, const float* __restrict__ bias,
            int M, int N, int K,
            long sAb, long sAh, int lda,
            long sBb, long sBh, int ldb,
            long sCb, long sCh, int ldc,
            float alpha, float beta, float diag_add)
{
  __shared__ float As[64][34];   // As[m][k]  (op(A) tile), 2 dwords row pad
  __shared__ float Bs[64][34];   // Bs[n][k]  (op(B)^T tile)
  int bh = blockIdx.z;
  const float* Ab = A + (long)(bh / NH) * sAb + (long)(bh % NH) * sAh;
  const float* Bb = B + (long)(bh / NH) * sBb + (long)(bh % NH) * sBh;
  float*       Cb = C + (long)(bh / NH) * sCb + (long)(bh % NH) * sCh;

  int tid  = threadIdx.x;
  int lane = tid & 31;
  int wave = tid >> 5;
  int m0t = blockIdx.y * 64, n0t = blockIdx.x * 64;
  int wr = (wave >> 1) * 32, wc = (wave & 1) * 32;
  int kq  = (lane >> 4) << 1;   // K offset within fragment
  int l16 = lane & 15;
  int hi  = lane >> 4;

  unsigned lds_A = (unsigned)(unsigned long long)(void*)&As[0][0];

  v8f z = {0.f,0.f,0.f,0.f,0.f,0.f,0.f,0.f};
  v8f acc[2][2] = {{z, z}, {z, z}};

  for (int k0 = 0; k0 < K; k0 += 32) {
    if (!TA) {
      // Async tensor DMA: 32 (K, contiguous) x 64 (M rows, stride lda) tile.
      if (wave == 0) {
        tdm_load_tile(lds_A, Ab + (long)m0t * lda + k0, lda, 32u, 64u,
                      /*interval 32 dwords*/4u, /*pad 2 dwords*/1u);
        __builtin_amdgcn_s_wait_tensorcnt(0);
      }
    } else {
      for (int idx = tid; idx < 2048; idx += 128) {
        int m = idx & 63, k = idx >> 6;
        int gm = m0t + m, gk = k0 + k;
        float v = 0.f;
        if (gm < M && gk < K) v = Ab[(long)gk * lda + gm];
        As[m][k] = v;
      }
    }
    if (TB) {
      for (int idx = tid; idx < 2048; idx += 128) {
        int k = idx & 31, n = idx >> 5;
        int gn = n0t + n, gk = k0 + k;
        float v = 0.f;
        if (gn < N && gk < K) v = Bb[(long)gn * ldb + gk];
        Bs[n][k] = v;
      }
    } else {
      for (int idx = tid; idx < 2048; idx += 128) {
        int n = idx & 63, k = idx >> 6;
        int gn = n0t + n, gk = k0 + k;
        float v = 0.f;
        if (gn < N && gk < K) v = Bb[(long)gk * ldb + gn];
        Bs[n][k] = v;
      }
    }
    __syncthreads();

#pragma unroll
    for (int kk = 0; kk < 32; kk += 4) {
      v2f a0 = *(const v2f*)&As[wr + l16][kk + kq];
      v2f a1 = *(const v2f*)&As[wr + 16 + l16][kk + kq];
      v2f b0 = *(const v2f*)&Bs[wc + l16][kk + kq];
      v2f b1 = *(const v2f*)&Bs[wc + 16 + l16][kk + kq];
      acc[0][0] = wmma4(a0, b0, acc[0][0]);
      acc[0][1] = wmma4(a0, b1, acc[0][1]);
      acc[1][0] = wmma4(a1, b0, acc[1][0]);
      acc[1][1] = wmma4(a1, b1, acc[1][1]);
    }
    __syncthreads();
  }

  // Epilogue.  C layout: lane holds (M = r + 8*(lane/16), N = lane%16) in vgpr r.
#pragma unroll
  for (int i = 0; i < 2; ++i) {
#pragma unroll
    for (int j = 0; j < 2; ++j) {
      int ng = n0t + wc + j * 16 + l16;
#pragma unroll
      for (int r = 0; r < 8; ++r) {
        int mg = m0t + wr + i * 16 + r + 8 * hi;
        if (mg < M && ng < N) {
          float val = alpha * acc[i][j][r];
          if (beta != 0.f) val += beta * Cb[(long)mg * ldc + ng];
          if (bias) val += bias[ng];
          if (diag_add != 0.f && mg == ng) val += diag_add;
          Cb[(long)mg * ldc + ng] = val;
        }
      }
    }
  }
}

// ---------------------------------------------------------------------------
// Per-(b,h) in-place Cholesky of a 512x512 matrix, right-looking, 1 WG each.
// Zeroes the strict upper triangle at the end.
// ---------------------------------------------------------------------------
__global__ __launch_bounds__(256)
void cholesky_k(float* A, long sbh, int N)
{
  float* Mx = A + (long)blockIdx.x * sbh;
  __shared__ float col[NSEQ];
  int t = threadIdx.x;
  for (int j = 0; j < N; ++j) {
    for (int i = j + t; i < N; i += 256) col[i] = Mx[(long)i * N + j];
    __syncthreads();
    float d = sqrtf(col[j]);
    float inv = 1.0f / d;
    __syncthreads();                      // all read col[j] before rewrite
    for (int i = j + t; i < N; i += 256) {
      float v = (i == j) ? d : col[i] * inv;
      Mx[(long)i * N + j] = v;
      col[i] = v;
    }
    __syncthreads();
    for (int i = j + 1 + t; i < N; i += 256) {
      float lij = col[i];
      float* row = Mx + (long)i * N;
      for (int k = j + 1; k <= i; ++k) row[k] -= lij * col[k];
    }
    __syncthreads();
  }
  for (int idx = t; idx < N * N; idx += 256) {
    int i = idx / N, k = idx % N;
    if (k > i) Mx[idx] = 0.f;
  }
}

// ---------------------------------------------------------------------------
// Blocked triangular solve, one thread per RHS column, no barriers.
// UPPER_T=false: solve L X = B (forward).  UPPER_T=true: solve L^T X = B.
// In-place (X == B) is safe for both.
// ---------------------------------------------------------------------------
template<bool UPPER_T>
__global__ __launch_bounds__(128)
void trsm_k(const float* __restrict__ L, const float* __restrict__ Bm,
            float* __restrict__ X,
            int N, int M, long sL, long sB, int ldb, long sX, int ldx)
{
  __shared__ float ys[128][65];
  int bh = blockIdx.y;
  const float* Lb = L + (long)bh * sL;
  const float* Bb = Bm + (long)bh * sB;
  float* Xb = X + (long)bh * sX;
  int c = blockIdx.x * 128 + threadIdx.x;
  if (c >= M) return;
  float* y = ys[threadIdx.x];

  if (!UPPER_T) {
    for (int jb = 0; jb < N; jb += 64) {
      for (int r = 0; r < 64; ++r) {
        int row = jb + r;
        const float* Lr = Lb + (long)row * N;
        float s = Bb[(long)row * ldb + c];
        for (int k = 0; k < jb; ++k) s -= Lr[k] * Xb[(long)k * ldx + c];
        y[r] = s;
      }
      for (int r = 0; r < 64; ++r) {
        int row = jb + r;
        const float* Lr = Lb + (long)row * N + jb;
        float s = y[r];
        for (int k = 0; k < r; ++k) s -= Lr[k] * y[k];
        s /= Lr[r];
        y[r] = s;
        Xb[(long)row * ldx + c] = s;
      }
    }
  } else {
    for (int jb = N - 64; jb >= 0; jb -= 64) {
      for (int r = 0; r < 64; ++r) {
        int row = jb + r;
        float s = Bb[(long)row * ldb + c];
        for (int k = jb + 64; k < N; ++k)
          s -= Lb[(long)k * N + row] * Xb[(long)k * ldx + c];
        y[r] = s;
      }
      for (int r = 63; r >= 0; --r) {
        int row = jb + r;
        float s = y[r];
        for (int k = r + 1; k < 64; ++k) s -= Lb[(long)(jb + k) * N + row] * y[k];
        s /= Lb[(long)row * N + row];
        y[r] = s;
        Xb[(long)row * ldx + c] = s;
      }
    }
  }
}

// ------------------------- elementwise helpers ------------------------------
__global__ void copy_k(float* dst, const float* src, long n) {
  for (long i = blockIdx.x * 256L + threadIdx.x; i < n; i += (long)gridDim.x * 256) dst[i] = src[i];
}
__global__ void copy_diag_k(float* dst, const float* src, long n, float dg) {
  for (long i = blockIdx.x * 256L + threadIdx.x; i < n; i += (long)gridDim.x * 256) {
    long ij = i % (long)(NSEQ * NSEQ);
    int r = (int)(ij / NSEQ), cc = (int)(ij % NSEQ);
    dst[i] = src[i] + ((r == cc) ? dg : 0.f);
  }
}
__global__ void add_k(float* dst, const float* src, long n) {
  for (long i = blockIdx.x * 256L + threadIdx.x; i < n; i += (long)gridDim.x * 256) dst[i] += src[i];
}
__global__ void rsub_k(float* dst, const float* a, long n) {   // dst = a - dst
  for (long i = blockIdx.x * 256L + threadIdx.x; i < n; i += (long)gridDim.x * 256) dst[i] = a[i] - dst[i];
}
// per-head squared norms: out[bh*NSEQ+s] = sum_vd X[b,s,h*64+vd]^2
__global__ void norms_k(const float* __restrict__ X, float* __restrict__ out) {
  long n = (long)NBH * NSEQ;
  for (long i = blockIdx.x * 256L + threadIdx.x; i < n; i += (long)gridDim.x * 256) {
    int bh = (int)(i / NSEQ), s = (int)(i % NSEQ);
    int b = bh / NH, h = bh % NH;
    const float* p = X + ((long)b * NSEQ + s) * HDIM + h * VDD;
    float acc = 0.f;
    for (int d = 0; d < VDD; ++d) { float v = p[d]; acc += v * v; }
    out[i] = acc;
  }
}
// gather [B,S,HD] head slice -> [BH,S,VD]
__global__ void gather_heads_k(const float* __restrict__ V, float* __restrict__ out) {
  long n = (long)NBH * NSEQ * VDD;
  for (long i = blockIdx.x * 256L + threadIdx.x; i < n; i += (long)gridDim.x * 256) {
    int vd = (int)(i % VDD);
    int s  = (int)((i / VDD) % NSEQ);
    int bh = (int)(i / ((long)VDD * NSEQ));
    int b = bh / NH, h = bh % NH;
    out[i] = V[((long)b * NSEQ + s) * HDIM + h * VDD + vd];
  }
}
// K[idx] currently = dot(a_i,b_j) -> sigma^2 * exp(-0.5*(na+nb-2dot)/VD)
__global__ void rbf_finish_k(float* __restrict__ Km, const float* __restrict__ na,
                             const float* __restrict__ nb, const float* sig, long n) {
  float sc = sig ? sig[0] * sig[0] : 1.f;
  for (long i = blockIdx.x * 256L + threadIdx.x; i < n; i += (long)gridDim.x * 256) {
    long rem = i % (long)(NSEQ * NSEQ);
    int bh = (int)(i / (long)(NSEQ * NSEQ));
    int r = (int)(rem / NSEQ), cc = (int)(rem % NSEQ);
    float d2 = na[(long)bh * NSEQ + r] + nb[(long)bh * NSEQ + cc] - 2.f * Km[i];
    Km[i] = sc * expf(-0.5f * d2 / (float)VDD);
  }
}
// T1[bh,s,ss*64+vd] = mean[bh,s,vd]
__global__ void fill_T1_k(const float* __restrict__ mean, float* __restrict__ T1) {
  long n = (long)NBH * NSEQ * SSAMP * VDD;
  for (long i = blockIdx.x * 256L + threadIdx.x; i < n; i += (long)gridDim.x * 256) {
    int t = (int)(i % (SSAMP * VDD));
    long bs = i / (SSAMP * VDD);           // bh*NSEQ + s
    T1[i] = mean[bs * VDD + (t & (VDD - 1))];
  }
}
// SF[((b*SS+ss)*S+s)*HD + h*64+vd] = T1[((b*H+h)*S+s)*256 + ss*64+vd]
__global__ void reshape_samp_k(const float* __restrict__ T1, float* __restrict__ SF) {
  long n = (long)NB * SSAMP * NSEQ * HDIM;
  for (long i = blockIdx.x * 256L + threadIdx.x; i < n; i += (long)gridDim.x * 256) {
    int hd = (int)(i % HDIM);
    int s  = (int)((i / HDIM) % NSEQ);
    int ss = (int)((i / ((long)HDIM * NSEQ)) % SSAMP);
    int b  = (int)(i / ((long)HDIM * NSEQ * SSAMP));
    int h = hd / VDD, vd = hd % VDD;
    SF[i] = T1[(((long)(b * NH + h) * NSEQ + s) * (SSAMP * VDD)) + ss * VDD + vd];
  }
}
// single-block final reduction -> scalar log-joint
__global__ __launch_bounds__(256)
void final_reduce_k(const float* __restrict__ QT, const float* __restrict__ KT,
                    const float* __restrict__ LQ, const float* __restrict__ LK,
                    float* __restrict__ out) {
  __shared__ float red[256];
  int t = threadIdx.x;
  float sq = 0.f, sk = 0.f, dq = 0.f, dk = 0.f;
  for (long i = t; i < (long)NBH * VDD * VDD; i += 256) { sq += QT[i]; sk += KT[i]; }
  for (int i = t; i < NBH * NSEQ; i += 256) {
    int bh = i / NSEQ, r = i % NSEQ;
    long d = (long)bh * NSEQ * NSEQ + (long)r * NSEQ + r;
    dq += logf(LQ[d]);
    dk += logf(LK[d]);
  }
  float res[4]; float in[4] = {sq, sk, dq, dk};
  for (int v = 0; v < 4; ++v) {
    red[t] = in[v]; __syncthreads();
    for (int s = 128; s > 0; s >>= 1) { if (t < s) red[t] += red[t + s]; __syncthreads(); }
    res[v] = red[0]; __syncthreads();
  }
  if (t == 0) {
    float lj = res[0] / (float)NB + 2.f * fabsf(res[2] / (float)NBH)
             + res[1] / (float)NB + 2.f * fabsf(res[3] / (float)NBH);
    out[0] = lj;
  }
}

// --------------------------- host-side driver -------------------------------
static inline void gemm(hipStream_t s, bool TA, bool TB,
                        const float* A, const float* B, float* C, const float* bias,
                        int M, int N, int K,
                        long sAb, long sAh, int lda,
                        long sBb, long sBh, int ldb,
                        long sCb, long sCh, int ldc,
                        float alpha, float beta, float diag, int nbh)
{
  dim3 g((N + 63) / 64, (M + 63) / 64, nbh), b(128);
  if (!TA && !TB) gemm_k<false, false><<<g, b, 0, s>>>(A, B, C, bias, M, N, K, sAb, sAh, lda, sBb, sBh, ldb, sCb, sCh, ldc, alpha, beta, diag);
  else if (!TA &&  TB) gemm_k<false, true ><<<g, b, 0, s>>>(A, B, C, bias, M, N, K, sAb, sAh, lda, sBb, sBh, ldb, sCb, sCh, ldc, alpha, beta, diag);
  else if ( TA && !TB) gemm_k<true,  false><<<g, b, 0, s>>>(A, B, C, bias, M, N, K, sAb, sAh, lda, sBb, sBh, ldb, sCb, sCh, ldc, alpha, beta, diag);
  else                 gemm_k<true,  true ><<<g, b, 0, s>>>(A, B, C, bias, M, N, K, sAb, sAh, lda, sBb, sBh, ldb, sCb, sCh, ldc, alpha, beta, diag);
}
static inline dim3 ewg(long n) { long b = (n + 255) / 256; if (b > 262144) b = 262144; return dim3((unsigned)b); }

extern "C" void kernel_launch(void* const* d_in, const int* in_sizes, int n_in,
                              void* d_out, int out_size, void* d_ws, size_t ws_size,
                              hipStream_t stream)
{
  (void)in_sizes; (void)n_in; (void)ws_size;
  const float* x   = (const float*)d_in[0];
  const float* Wq  = (const float*)d_in[1];
  const float* Wk  = (const float*)d_in[2];
  const float* Wv  = (const float*)d_in[3];
  const float* Wx0 = (const float*)d_in[4];
  const float* Wo  = (const float*)d_in[5];
  const float* bo  = (const float*)d_in[6];
  const float* sq  = (const float*)d_in[7];
  const float* sk  = (const float*)d_in[8];
  const float* ez0 = (const float*)d_in[9];
  const float* eps = (const float*)d_in[10];
  float* out = (float*)d_out;

  const long n2   = (long)NSEQ * NSEQ;                 // 262144
  const long BIGN = (long)NBH * n2;                    // 8,388,608 floats (32MB)
  const long SM   = (long)NBH * NSEQ * VDD;            // 1,048,576 floats (4MB)
  const long BSH  = (long)NB * NSEQ * HDIM;            // 1,048,576 floats

  float* w = (float*)d_ws;
  size_t off = 0;
  auto alloc = [&](long n) { float* p = w + off; off += (size_t)n; return p; };

  float* Bs[10]; for (int i = 0; i < 10; ++i) Bs[i] = alloc(BIGN);
  float* Q  = alloc(BSH); float* Kx = alloc(BSH); float* V = alloc(BSH); float* X0 = alloc(BSH);
  float* FK = alloc(SM);  float* Z0 = alloc(SM);  float* T0 = alloc(SM);
  float* AFK = alloc(SM); float* MEAN = alloc(SM); float* MQ = alloc(SM); float* MK = alloc(SM);
  float* NQ = alloc((long)NBH * NSEQ); float* NK = alloc((long)NBH * NSEQ); float* NX = alloc((long)NBH * NSEQ);
  float* QT = alloc((long)NBH * VDD * VDD); float* KT = alloc((long)NBH * VDD * VDD);

  // strides for common views
  const long sKb = (long)NH * n2, sKh = n2;                       // [BH,S,S]
  const long sHb = (long)NSEQ * HDIM, sHh = VDD;                  // head slice of [B,S,HD]
  const long sSb = (long)NH * NSEQ * VDD, sSh = (long)NSEQ * VDD; // [BH,S,VD]

  // (a) projections  q,k,v,x0 : [B,S,HD] = x @ W^T
  gemm(stream, false, true, x, Wq,  Q,  nullptr, NB*NSEQ, HDIM, HDIM, 0,0,HDIM, 0,0,HDIM, 0,0,HDIM, 1.f,0.f,0.f, 1);
  gemm(stream, false, true, x, Wk,  Kx, nullptr, NB*NSEQ, HDIM, HDIM, 0,0,HDIM, 0,0,HDIM, 0,0,HDIM, 1.f,0.f,0.f, 1);
  gemm(stream, false, true, x, Wv,  V,  nullptr, NB*NSEQ, HDIM, HDIM, 0,0,HDIM, 0,0,HDIM, 0,0,HDIM, 1.f,0.f,0.f, 1);
  gemm(stream, false, true, x, Wx0, X0, nullptr, NB*NSEQ, HDIM, HDIM, 0,0,HDIM, 0,0,HDIM, 0,0,HDIM, 1.f,0.f,0.f, 1);

  // (b) per-head squared norms
  norms_k<<<ewg((long)NBH*NSEQ), 256, 0, stream>>>(Q,  NQ);
  norms_k<<<ewg((long)NBH*NSEQ), 256, 0, stream>>>(Kx, NK);
  norms_k<<<ewg((long)NBH*NSEQ), 256, 0, stream>>>(X0, NX);

  // (c) RBF kernel matrices: dot via WMMA gemm (NT, K=64) then exp-finish
  struct { const float* a; const float* b; float* dst; const float* na; const float* nb; const float* sig; } R[7] = {
    {Q,  Q,  Bs[1], NQ, NQ, sq},      // Kqq
    {Kx, Kx, Bs[2], NK, NK, sk},      // Kkk
    {X0, X0, Bs[0], NX, NX, nullptr}, // K0
    {Q,  X0, Bs[3], NQ, NX, nullptr}, // Kqx0
    {X0, Q,  Bs[4], NX, NQ, nullptr}, // Kx0q
    {X0, Kx, Bs[5], NX, NK, nullptr}, // Kx0k
    {Kx, X0, Bs[6], NK, NX, nullptr}, // Kkx0
  };
  for (int r = 0; r < 7; ++r) {
    gemm(stream, false, true, R[r].a, R[r].b, R[r].dst, nullptr, NSEQ, NSEQ, VDD,
         sHb, sHh, HDIM, sHb, sHh, HDIM, sKb, sKh, NSEQ, 1.f, 0.f, 0.f, NBH);
    rbf_finish_k<<<ewg(BIGN), 256, 0, stream>>>(R[r].dst, R[r].na, R[r].nb, R[r].sig, BIGN);
  }

  // (d) Cholesky factors L0 (Bs[7]) and Lkk (Bs[8])
  copy_diag_k<<<ewg(BIGN), 256, 0, stream>>>(Bs[7], Bs[0], BIGN, JITV);
  cholesky_k<<<NBH, 256, 0, stream>>>(Bs[7], n2, NSEQ);
  copy_diag_k<<<ewg(BIGN), 256, 0, stream>>>(Bs[8], Bs[2], BIGN, JITV);
  cholesky_k<<<NBH, 256, 0, stream>>>(Bs[8], n2, NSEQ);

  // (e) f_K = (Kkk + jit I) @ v  -> FK = Kkk@v + jit*v
  gather_heads_k<<<ewg(SM), 256, 0, stream>>>(V, FK);
  gemm(stream, false, false, Bs[2], V, FK, nullptr, NSEQ, VDD, NSEQ,
       sKb, sKh, NSEQ, sHb, sHh, HDIM, sSb, sSh, VDD, 1.f, JITV, 0.f, NBH);

  // (f) z0 = L0 @ eps_z0
  gemm(stream, false, false, Bs[7], ez0, Z0, nullptr, NSEQ, VDD, NSEQ,
       sKb, sKh, NSEQ, sSb, sSh, VDD, sSb, sSh, VDD, 1.f, 0.f, 0.f, NBH);

  // (g) v0 = Lkk^{-1} Kx0k  -> Bs[9]
  trsm_k<false><<<dim3(4, NBH), 128, 0, stream>>>(Bs[8], Bs[5], Bs[9], NSEQ, NSEQ, n2, n2, NSEQ, n2, NSEQ);

  // (h) Afk = Kx0k @ (Kkk^{-1} f_K)
  trsm_k<false><<<dim3(1, NBH), 128, 0, stream>>>(Bs[8], FK, T0, NSEQ, VDD, n2, NSEQ*VDD, VDD, NSEQ*VDD, VDD);
  trsm_k<true ><<<dim3(1, NBH), 128, 0, stream>>>(Bs[8], T0, T0, NSEQ, VDD, n2, NSEQ*VDD, VDD, NSEQ*VDD, VDD);
  gemm(stream, false, false, Bs[5], T0, AFK, nullptr, NSEQ, VDD, NSEQ,
       sKb, sKh, NSEQ, sSb, sSh, VDD, sSb, sSh, VDD, 1.f, 0.f, 0.f, NBH);

  // (i) E = K0 - v0^T v0 + Afk Afk^T   (accumulated in place over K0 = Bs[0])
  gemm(stream, true, false, Bs[9], Bs[9], Bs[0], nullptr, NSEQ, NSEQ, NSEQ,
       sKb, sKh, NSEQ, sKb, sKh, NSEQ, sKb, sKh, NSEQ, -1.f, 1.f, 0.f, NBH);
  gemm(stream, false, true, AFK, AFK, Bs[0], nullptr, NSEQ, NSEQ, VDD,
       sSb, sSh, VDD, sSb, sSh, VDD, sKb, sKh, NSEQ, 1.f, 1.f, 0.f, NBH);

  // (j) v1 = L0^{-1} Kx0q  (in place -> Bs[4])
  trsm_k<false><<<dim3(4, NBH), 128, 0, stream>>>(Bs[7], Bs[4], Bs[4], NSEQ, NSEQ, n2, n2, NSEQ, n2, NSEQ);
  // (k) BmT = L0^{-T} v1 -> Bs[8]  (Lkk no longer needed)
  trsm_k<true ><<<dim3(4, NBH), 128, 0, stream>>>(Bs[7], Bs[4], Bs[8], NSEQ, NSEQ, n2, n2, NSEQ, n2, NSEQ);
  // (l) G1 = BmT^T @ E -> Bs[9] ;  (m) G2 = G1 @ BmT -> Bs[0]
  gemm(stream, true, false, Bs[8], Bs[0], Bs[9], nullptr, NSEQ, NSEQ, NSEQ,
       sKb, sKh, NSEQ, sKb, sKh, NSEQ, sKb, sKh, NSEQ, 1.f, 0.f, 0.f, NBH);
  gemm(stream, false, false, Bs[9], Bs[8], Bs[0], nullptr, NSEQ, NSEQ, NSEQ,
       sKb, sKh, NSEQ, sKb, sKh, NSEQ, sKb, sKh, NSEQ, 1.f, 0.f, 0.f, NBH);

  // (n) vk = L0^{-1} Kx0k (in place -> Bs[5]) ; (o) w1 = L0^{-T} vk -> Bs[9]
  trsm_k<false><<<dim3(4, NBH), 128, 0, stream>>>(Bs[7], Bs[5], Bs[5], NSEQ, NSEQ, n2, n2, NSEQ, n2, NSEQ);
  copy_k<<<ewg(BIGN), 256, 0, stream>>>(Bs[9], Bs[5], BIGN);
  trsm_k<true ><<<dim3(4, NBH), 128, 0, stream>>>(Bs[7], Bs[9], Bs[9], NSEQ, NSEQ, n2, n2, NSEQ, n2, NSEQ);
  // (p) Kqk = Kqx0 @ w1 -> Bs[8] ; (q) mean = Kqk @ v
  gemm(stream, false, false, Bs[3], Bs[9], Bs[8], nullptr, NSEQ, NSEQ, NSEQ,
       sKb, sKh, NSEQ, sKb, sKh, NSEQ, sKb, sKh, NSEQ, 1.f, 0.f, 0.f, NBH);
  gemm(stream, false, false, Bs[8], V, MEAN, nullptr, NSEQ, VDD, NSEQ,
       sKb, sKh, NSEQ, sHb, sHh, HDIM, sSb, sSh, VDD, 1.f, 0.f, 0.f, NBH);

  // (r) cq = Kqq - v1^T v1 -> Bs[9]
  copy_k<<<ewg(BIGN), 256, 0, stream>>>(Bs[9], Bs[1], BIGN);
  gemm(stream, true, false, Bs[4], Bs[4], Bs[9], nullptr, NSEQ, NSEQ, NSEQ,
       sKb, sKh, NSEQ, sKb, sKh, NSEQ, sKb, sKh, NSEQ, -1.f, 1.f, 0.f, NBH);

  // (s) Lc = chol(cq + G2 - mean mean^T + jit I) -> Bs[4]
  copy_k<<<ewg(BIGN), 256, 0, stream>>>(Bs[4], Bs[9], BIGN);
  add_k<<<ewg(BIGN), 256, 0, stream>>>(Bs[4], Bs[0], BIGN);
  gemm(stream, false, true, MEAN, MEAN, Bs[4], nullptr, NSEQ, NSEQ, VDD,
       sSb, sSh, VDD, sSb, sSh, VDD, sKb, sKh, NSEQ, -1.f, 1.f, JITV, NBH);
  cholesky_k<<<NBH, 256, 0, stream>>>(Bs[4], n2, NSEQ);

  // (t) ck = Kkk - vk^T vk -> Bs[0]
  copy_k<<<ewg(BIGN), 256, 0, stream>>>(Bs[0], Bs[2], BIGN);
  gemm(stream, true, false, Bs[5], Bs[5], Bs[0], nullptr, NSEQ, NSEQ, NSEQ,
       sKb, sKh, NSEQ, sKb, sKh, NSEQ, sKb, sKh, NSEQ, -1.f, 1.f, 0.f, NBH);

  // (u) Lq_chol = chol(cq + jit I) -> Bs[5] ; (v) Lk_chol = chol(ck + jit I) -> Bs[2]
  copy_diag_k<<<ewg(BIGN), 256, 0, stream>>>(Bs[5], Bs[9], BIGN, JITV);
  cholesky_k<<<NBH, 256, 0, stream>>>(Bs[5], n2, NSEQ);
  copy_diag_k<<<ewg(BIGN), 256, 0, stream>>>(Bs[2], Bs[0], BIGN, JITV);
  cholesky_k<<<NBH, 256, 0, stream>>>(Bs[2], n2, NSEQ);

  // (w) w2 = K0^{-1} z0 (in place on Z0) ; mq, mk
  trsm_k<false><<<dim3(1, NBH), 128, 0, stream>>>(Bs[7], Z0, Z0, NSEQ, VDD, n2, NSEQ*VDD, VDD, NSEQ*VDD, VDD);
  trsm_k<true ><<<dim3(1, NBH), 128, 0, stream>>>(Bs[7], Z0, Z0, NSEQ, VDD, n2, NSEQ*VDD, VDD, NSEQ*VDD, VDD);
  gemm(stream, false, false, Bs[3], Z0, MQ, nullptr, NSEQ, VDD, NSEQ,
       sKb, sKh, NSEQ, sSb, sSh, VDD, sSb, sSh, VDD, 1.f, 0.f, 0.f, NBH);
  gemm(stream, false, false, Bs[6], Z0, MK, nullptr, NSEQ, VDD, NSEQ,
       sKb, sKh, NSEQ, sSb, sSh, VDD, sSb, sSh, VDD, 1.f, 0.f, 0.f, NBH);

  // (x) Lq = Lq_chol^{-1}(mean - mq); qt = Lq^T Lq.  Same for k with f_K - mk.
  rsub_k<<<ewg(SM), 256, 0, stream>>>(MQ, MEAN, SM);
  trsm_k<false><<<dim3(1, NBH), 128, 0, stream>>>(Bs[5], MQ, MQ, NSEQ, VDD, n2, NSEQ*VDD, VDD, NSEQ*VDD, VDD);
  gemm(stream, true, false, MQ, MQ, QT, nullptr, VDD, VDD, NSEQ,
       sSb, sSh, VDD, sSb, sSh, VDD, (long)NH*VDD*VDD, (long)VDD*VDD, VDD, 1.f, 0.f, 0.f, NBH);
  rsub_k<<<ewg(SM), 256, 0, stream>>>(MK, FK, SM);
  trsm_k<false><<<dim3(1, NBH), 128, 0, stream>>>(Bs[2], MK, MK, NSEQ, VDD, n2, NSEQ*VDD, VDD, NSEQ*VDD, VDD);
  gemm(stream, true, false, MK, MK, KT, nullptr, VDD, VDD, NSEQ,
       sSb, sSh, VDD, sSb, sSh, VDD, (long)NH*VDD*VDD, (long)VDD*VDD, VDD, 1.f, 0.f, 0.f, NBH);

  // (y) samples: T1 = mean (bcast over SS) + Lc @ eps^T ; reshape ; @Wo^T + bo
  float* T1 = Bs[7];  // L0 retired
  float* SF = Bs[0];  // ck retired
  const long eN = (long)SSAMP * VDD;                 // 256 columns
  fill_T1_k<<<ewg((long)NBH * NSEQ * eN), 256, 0, stream>>>(MEAN, T1);
  gemm(stream, false, true, Bs[4], eps, T1, nullptr, NSEQ, (int)eN, NSEQ,
       sKb, sKh, NSEQ, (long)NH * eN * NSEQ, eN * NSEQ, NSEQ,
       (long)NH * NSEQ * eN, (long)NSEQ * eN, (int)eN, 1.f, 1.f, 0.f, NBH);
  reshape_samp_k<<<ewg((long)NB * SSAMP * NSEQ * HDIM), 256, 0, stream>>>(T1, SF);
  gemm(stream, false, true, SF, Wo, out, bo, NB * SSAMP * NSEQ, HDIM, HDIM,
       0, 0, HDIM, 0, 0, HDIM, 0, 0, HDIM, 1.f, 0.f, 0.f, 1);

  // (z) scalar log-joint -> out[out_size-1]
  final_reduce_k<<<1, 256, 0, stream>>>(QT, KT, Bs[5], Bs[2], out + (out_size - 1));
}